// ExtSummarizer_32298154065982
// MI455X (gfx1250) — compile-verified
//
#include <hip/hip_runtime.h>
#include <hip/hip_bf16.h>

typedef __attribute__((ext_vector_type(2))) float v2f;
typedef __attribute__((ext_vector_type(8))) float v8f;

#define B_ 512
#define S_ 100
#define H_ 768
#define LAMB 0.8f

#define WMMA_F32(a, b, c) \
  __builtin_amdgcn_wmma_f32_16x16x4_f32(false, (a), false, (b), (short)0, (c), false, false)

// ---------------------------------------------------------------- d_rep[b,h]
__global__ __launch_bounds__(256) void k_drep(const float* __restrict__ X,
                                              float* __restrict__ drep) {
  int idx = blockIdx.x * 256 + threadIdx.x;           // b*H + h
  if (idx >= B_ * H_) return;
  int b = idx / H_, h = idx % H_;
  const float* p = X + (size_t)b * S_ * H_ + h;
  float s = 0.f;
  for (int i = 0; i < S_; ++i) s += p[(size_t)i * H_];
  drep[idx] = s * (1.0f / S_);
}

// ------------------------------------------- w[b,h] = sum_k Wrel[h,k]*drep[b,k]
__global__ __launch_bounds__(256) void k_wrel(const float* __restrict__ Wrel,
                                              const float* __restrict__ drep,
                                              float* __restrict__ w) {
  __shared__ float ds[H_];
  int b = blockIdx.x;
  for (int k = threadIdx.x; k < H_; k += 256) ds[k] = drep[b * H_ + k];
  __syncthreads();
  for (int h = threadIdx.x; h < H_; h += 256) {
    const float* row = Wrel + (size_t)h * H_;
    float s = 0.f;
    for (int k = 0; k < H_; ++k) s += row[k] * ds[k];
    w[b * H_ + h] = s;
  }
}

// ------------------- cont[b,i] = x . Wcont ; rel[b,i] = x . w[b]  (wave/row)
__global__ __launch_bounds__(256) void k_contrel(const float* __restrict__ X,
                                                 const float* __restrict__ Wcont,
                                                 const float* __restrict__ w,
                                                 float* __restrict__ cont,
                                                 float* __restrict__ rel) {
  int wave = threadIdx.x >> 5, lane = threadIdx.x & 31;
  int r = blockIdx.x * 8 + wave;                      // row in [0, B*S)
  if (r >= B_ * S_) return;
  int b = r / S_;
  const float* xr = X + (size_t)r * H_;
  const float* wb = w + (size_t)b * H_;
  float ac = 0.f, ar = 0.f;
  for (int h = lane; h < H_; h += 32) {
    float x = xr[h];
    ac += x * Wcont[h];
    ar += x * wb[h];
  }
  for (int off = 16; off; off >>= 1) {
    ac += __shfl_xor(ac, off, 32);
    ar += __shfl_xor(ar, off, 32);
  }
  if (lane == 0) { cont[r] = ac; rel[r] = ar; }
}

// ---- Pack Wsim into K-pair-major: Wp_v2f[kpair*H + n] = {W[2k][n], W[2k+1][n]}
// so each WMMA B fragment is one aligned b64 load.
__global__ __launch_bounds__(256) void k_packB(const float* __restrict__ W,
                                               float* __restrict__ Wp) {
  int idx = blockIdx.x * 256 + threadIdx.x;           // k*H + n
  if (idx >= H_ * H_) return;
  int k = idx / H_, n = idx % H_;
  Wp[((size_t)(k >> 1) * H_ + n) * 2 + (k & 1)] = W[idx];
}

// --------------------------- GEMM1: U[51200,768] = X[51200,768] * Wsim[768,768]
// fp32 WMMA 16x16x4; 32x32 register block per wave (4 accumulators) ->
// per K-step: 2 b64 A + 2 b64 B feed 4 WMMAs (1 VMEM per WMMA).
__global__ __launch_bounds__(256) void k_gemm1(const float* __restrict__ X,
                                               const float* __restrict__ Wp,
                                               float* __restrict__ U) {
  const int lane = threadIdx.x & 31;
  const int wave = threadIdx.x >> 5;
  const int lo = lane & 15;          // A: M row / B,D: N col
  const int hi = lane >> 4;          // K-pair select / D: M high bit
  const int n0 = blockIdx.x * 32;                       // 24 N-blocks
  const int row0 = (blockIdx.y * 8 + wave) * 32;        // 1600 M-blocks
  const float* A0 = X + (size_t)(row0 + lo) * H_;
  const float* A1 = X + (size_t)(row0 + 16 + lo) * H_;
  const v2f* Wv = (const v2f*)Wp;
  v8f acc00 = {}, acc01 = {}, acc10 = {}, acc11 = {};
  for (int k0 = 0; k0 < H_; k0 += 4) {
    int ka = k0 + 2 * hi;
    v2f a0 = *(const v2f*)(A0 + ka);                    // A[M=lo   ][K=ka,ka+1]
    v2f a1 = *(const v2f*)(A1 + ka);                    // A[M=16+lo][K=ka,ka+1]
    const v2f* bp = Wv + (size_t)((k0 >> 1) + hi) * H_ + n0 + lo;
    v2f b0 = bp[0];                                     // B[K pair][N=lo]
    v2f b1 = bp[16];                                    // B[K pair][N=16+lo]
    acc00 = WMMA_F32(a0, b0, acc00);
    acc01 = WMMA_F32(a0, b1, acc01);
    acc10 = WMMA_F32(a1, b0, acc10);
    acc11 = WMMA_F32(a1, b1, acc11);
  }
#pragma unroll
  for (int r = 0; r < 8; ++r) {
    size_t r0 = (size_t)(row0 + r + 8 * hi) * H_;
    size_t r1 = (size_t)(row0 + 16 + r + 8 * hi) * H_;
    U[r0 + n0 + lo]      = acc00[r];
    U[r0 + n0 + 16 + lo] = acc01[r];
    U[r1 + n0 + lo]      = acc10[r];
    U[r1 + n0 + 16 + lo] = acc11[r];
  }
}

// --------- GEMM2 + epilogue: q[b,i,j] = sigmoid(rel*mij + cont + sim*mij + bias)*mij
// sim[b] = U_b[100,768] * X_b[100,768]^T ; 7x7 padded 16-tiles, masked store.
__global__ __launch_bounds__(224) void k_gemm2(const float* __restrict__ X,
                                               const float* __restrict__ U,
                                               const float* __restrict__ cont,
                                               const float* __restrict__ rel,
                                               const int* __restrict__ mask,
                                               const float* __restrict__ bptr,
                                               float* __restrict__ q) {
  const int lane = threadIdx.x & 31;
  const int wave = threadIdx.x >> 5;                    // 0..6 -> i-tile
  const int lo = lane & 15, hi = lane >> 4;
  const int b = blockIdx.y;
  const int j0 = blockIdx.x * 16;
  const int i0 = wave * 16;
  const int ar = i0 + lo;  const int arc = ar < S_ ? ar : S_ - 1;
  const int br = j0 + lo;  const int brc = br < S_ ? br : S_ - 1;
  const float* Ap = U + ((size_t)(b * S_ + arc)) * H_;  // A[M=lo][K]
  const float* Bp = X + ((size_t)(b * S_ + brc)) * H_;  // B[K][N=lo] = X[j][k]
  v8f acc = {};
  for (int k0 = 0; k0 < H_; k0 += 4) {
    int ka = k0 + 2 * hi;
    v2f a  = *(const v2f*)(Ap + ka);
    v2f bb = *(const v2f*)(Bp + ka);
    acc = WMMA_F32(a, bb, acc);
  }
  float bias = bptr[0];
  int j = j0 + lo;
  if (j < S_) {
    float mj = (float)mask[b * S_ + j];
#pragma unroll
    for (int r = 0; r < 8; ++r) {
      int i = i0 + r + 8 * hi;
      if (i < S_) {
        float mi  = (float)mask[b * S_ + i];
        float mij = mi * mj;
        float x = rel[b * S_ + i] * mij + cont[b * S_ + i] + acc[r] * mij + bias;
        float s = mij / (1.0f + __expf(-x));
        q[((size_t)b * S_ + i) * S_ + j] = s;
      }
    }
  }
}

// ----------------------------------------------- sum_vec[b,j] = sum_i q[b,i,j]
__global__ __launch_bounds__(256) void k_colsum(const float* __restrict__ q,
                                                float* __restrict__ sv) {
  int idx = blockIdx.x * 256 + threadIdx.x;             // b*S + j
  if (idx >= B_ * S_) return;
  int b = idx / S_, j = idx % S_;
  const float* qb = q + (size_t)b * S_ * S_ + j;
  float s = 0.f;
  for (int i = 0; i < S_; ++i) s += qb[(size_t)i * S_];
  sv[idx] = s;
}

// --- Neumann solve: y = y0 + (lamb*qD) y, 100 iters; score = (1-lamb)*y.
// ||lamb*qD||_1 = 0.8 -> 0.8^100 ~ 2e-10 < fp32 eps. A in LDS, stride 101.
__global__ __launch_bounds__(128) void k_solve(const float* __restrict__ q,
                                               const float* __restrict__ svec,
                                               float* __restrict__ out) {
  __shared__ float As[S_ * 101];
  __shared__ float sv[S_];
  __shared__ float ys[S_];
  int b = blockIdx.x, tid = threadIdx.x;
  for (int j = tid; j < S_; j += 128) sv[j] = LAMB / svec[b * S_ + j];
  __syncthreads();
  const float* qb = q + (size_t)b * S_ * S_;
  for (int t = tid; t < S_ * S_; t += 128) {
    int i = t / S_, j = t % S_;
    As[i * 101 + j] = qb[t] * sv[j];
  }
  if (tid < S_) ys[tid] = 1.0f / S_;
  __syncthreads();
  for (int it = 0; it < 100; ++it) {
    float acc = 1.0f / S_;
    if (tid < S_) {
      const float* row = As + tid * 101;
      for (int j = 0; j < S_; ++j) acc += row[j] * ys[j];
    }
    __syncthreads();
    if (tid < S_) ys[tid] = acc;
    __syncthreads();
  }
  if (tid < S_) out[b * S_ + tid] = (1.0f - LAMB) * ys[tid];
}

extern "C" void kernel_launch(void* const* d_in, const int* in_sizes, int n_in,
                              void* d_out, int out_size, void* d_ws, size_t ws_size,
                              hipStream_t stream) {
  const float* X     = (const float*)d_in[0];   // [B,S,H]
  const int*   mask  = (const int*)  d_in[1];   // [B,S]
  const float* Wcont = (const float*)d_in[2];   // [1,H]
  const float* Wsim  = (const float*)d_in[3];   // [H,H]
  const float* Wrel  = (const float*)d_in[4];   // [H,H]
  const float* bmat  = (const float*)d_in[5];   // [1,1]
  float* out = (float*)d_out;                   // [B,S]

  // workspace carve-up (floats)
  float* ws   = (float*)d_ws;
  float* U    = ws;                         size_t o = (size_t)B_ * S_ * H_; // 39.3M
  float* q    = ws + o;                     o += (size_t)B_ * S_ * S_;      // 5.12M
  float* Wp   = ws + o;                     o += (size_t)H_ * H_;           // packed Wsim
  float* drep = ws + o;                     o += (size_t)B_ * H_;
  float* w    = ws + o;                     o += (size_t)B_ * H_;
  float* cont = ws + o;                     o += (size_t)B_ * S_;
  float* rel  = ws + o;                     o += (size_t)B_ * S_;
  float* svec = ws + o;                     o += (size_t)B_ * S_;

  k_drep   <<<(B_ * H_ + 255) / 256, 256, 0, stream>>>(X, drep);
  k_wrel   <<<B_, 256, 0, stream>>>(Wrel, drep, w);
  k_contrel<<<(B_ * S_) / 8, 256, 0, stream>>>(X, Wcont, w, cont, rel);
  k_packB  <<<(H_ * H_ + 255) / 256, 256, 0, stream>>>(Wsim, Wp);
  k_gemm1  <<<dim3(H_ / 32, (B_ * S_) / (32 * 8)), 256, 0, stream>>>(X, Wp, U);
  k_gemm2  <<<dim3(7, B_), 224, 0, stream>>>(X, U, cont, rel, mask, bmat, q);
  k_colsum <<<(B_ * S_ + 255) / 256, 256, 0, stream>>>(q, svec);
  k_solve  <<<B_, 128, 0, stream>>>(q, svec, out);
}